// NCC_3994319586063
// MI455X (gfx1250) — compile-verified
//
#include <hip/hip_runtime.h>
#include <hip/hip_bf16.h>

// NCC 3D loss for MI455X (gfx1250, wave32).
// Pass 1: fused W-axis 11-tap box sums of (I, J, I*I, J*J, I*J)  -> ws buf0
//         (LDS staging via GLOBAL_LOAD_ASYNC_TO_LDS_B32, ASYNCcnt-tracked)
// Pass 2: H-axis 11-tap box sums via V_WMMA_F32_16X16X4_F32 band-matrix
//         products on 16x16 tiles; constant-offset loads off one base ptr.
// Pass 3: D-axis running-window sums + cc math + global reduction -> d_out
//
// Intermediate layout: plane-interleaved [d][field][h][w] so the five fields
// of one plane sit within 1.25 MB (=> 24-bit immediate load offsets).
//
// Workspace requirement: 2 * 5 * 192*256*256 * 4 bytes = 503,316,480 bytes.

typedef __attribute__((ext_vector_type(2))) float v2f;
typedef __attribute__((ext_vector_type(8))) float v8f;

#define D_   192
#define H_   256
#define W_   256
#define HW_  (H_ * W_)           // 65536
#define NVOX (D_ * HW_)          // 12582912
#define PAD  5
#define INV_VOL (1.0f / 1331.0f)
#define EPS_ 1e-5f
#define DCH  48                  // D-chunk per thread in pass 3 (192/48 = 4)

#if defined(__has_builtin)
#if __has_builtin(__builtin_amdgcn_global_load_async_to_lds_b32)
#define HAVE_ASYNC_LDS 1
#endif
#endif
#ifndef HAVE_ASYNC_LDS
#define HAVE_ASYNC_LDS 0
#endif

// Builtin signature (from hipcc diagnostic): (as1 int* gsrc, as3 int* lds, Ii, Ii)
typedef __attribute__((address_space(1))) int as1_int;
typedef __attribute__((address_space(3))) int as3_int;
#define AS1I(p) ((as1_int*)(p))
#define AS3I(p) ((as3_int*)(p))

#if defined(__has_builtin) && __has_builtin(__builtin_amdgcn_s_wait_asynccnt)
#define WAIT_ASYNC() __builtin_amdgcn_s_wait_asynccnt(0)
#else
#define WAIT_ASYNC() asm volatile("s_wait_asynccnt 0x0" ::: "memory")
#endif

// ---------------------------------------------------------------------------
// Pass 1: one block per (d,h) row; W-axis windowed sums of 5 quantities.
// ---------------------------------------------------------------------------
__global__ void ncc_wpass(const float* __restrict__ I, const float* __restrict__ J,
                          float* __restrict__ out5, float* __restrict__ out_scalar) {
    __shared__ float sI[W_];
    __shared__ float sJ[W_];
    const int w = threadIdx.x;                    // 0..255
    const int d = blockIdx.x >> 8;
    const int h = blockIdx.x & (H_ - 1);
    const size_t ibase = (size_t)blockIdx.x * W_; // I/J are dense [d][h][w]

#if HAVE_ASYNC_LDS
    // CDNA5 async global->LDS copies (ASYNCcnt-tracked, bypass VGPRs).
    __builtin_amdgcn_global_load_async_to_lds_b32(AS1I(I + ibase + w), AS3I(&sI[w]), 0, 0);
    __builtin_amdgcn_global_load_async_to_lds_b32(AS1I(J + ibase + w), AS3I(&sJ[w]), 0, 0);
    WAIT_ASYNC();
#else
    sI[w] = I[ibase + w];
    sJ[w] = J[ibase + w];
#endif
    __syncthreads();

    float s0 = 0.f, s1 = 0.f, s2 = 0.f, s3 = 0.f, s4 = 0.f;
#pragma unroll
    for (int k = -PAD; k <= PAD; ++k) {
        int x = w + k;
        if (0 <= x && x < W_) {
            float a = sI[x], b = sJ[x];
            s0 += a;      s1 += b;
            s2 += a * a;  s3 += b * b;  s4 += a * b;
        }
    }
    float* o = out5 + (size_t)d * (5 * HW_) + h * W_ + w;
    o[0 * HW_] = s0;
    o[1 * HW_] = s1;
    o[2 * HW_] = s2;
    o[3 * HW_] = s3;
    o[4 * HW_] = s4;

    if (blockIdx.x == 0 && threadIdx.x == 0) *out_scalar = 0.f;  // before pass 3 atomics
}

// ---------------------------------------------------------------------------
// Pass 2: H-axis box sums via WMMA. One wave (32 lanes) per 16x16 (h,w) tile
// of one field at one depth slice. out_tile = sum_p Band_p(16x16) * T_p(16x16),
// each product decomposed into chained V_WMMA_F32_16X16X4_F32; all-zero band
// chunks are skipped statically (8 WMMAs per tile). All data loads/stores use
// one per-lane base pointer + compile-time immediate offsets (row stride 1KB).
// ---------------------------------------------------------------------------
__global__ void ncc_hpass_wmma(const float* __restrict__ in5, float* __restrict__ out5) {
    const int lane   = threadIdx.x & 31;
    const int waveid = threadIdx.x >> 5;                 // 0..7
    const int task   = blockIdx.x * 8 + waveid;          // < 5*192*256

    const int wt   = task & 15;
    const int ht   = (task >> 4) & 15;
    const int rest = task >> 8;                          // arr*192 + d
    const int d    = rest % D_;
    const int arr  = rest / D_;

    const float* in  = in5  + ((size_t)d * 5 + arr) * HW_;
    float*       out = out5 + ((size_t)d * 5 + arr) * HW_;

    const int half = lane >> 4;     // 0: K in {0,1}; 1: K in {2,3}
    const int m    = lane & 15;     // A row index / B,D column index
    const int h0   = ht * 16;
    const int w0   = wt * 16;

    // Single per-lane base pointers; every access below is base + constant.
    const float* pb = in  + (size_t)((h0 + 2 * half) * W_ + w0 + m);
    float*       ob = out + (size_t)((h0 + 8 * half) * W_ + w0 + m);

    v8f acc = {};

    // One K=4 chunk: JB0 = 16*p + 4*c (constant). Band A[i][j] = (|16p+j - i|<=5),
    // exact 0/1 in fp32; data rows h0+JB0+2*half (+1) at column w0+m.
#define NCC_CHUNK(JB0)                                                              \
    {                                                                               \
        v2f a, b;                                                                   \
        a.x = ((unsigned)((JB0) + 2 * half     - m + PAD) <= 2u * PAD) ? 1.f : 0.f; \
        a.y = ((unsigned)((JB0) + 2 * half + 1 - m + PAD) <= 2u * PAD) ? 1.f : 0.f; \
        b.x = pb[((JB0)    ) * W_];                                                 \
        b.y = pb[((JB0) + 1) * W_];                                                 \
        acc = __builtin_amdgcn_wmma_f32_16x16x4_f32(                                \
            /*neg_a=*/false, a, /*neg_b=*/false, b,                                 \
            /*c_mod=*/(short)0, acc, /*reuse_a=*/false, /*reuse_b=*/false);         \
    }

    if (ht > 0)  { NCC_CHUNK(-8) NCC_CHUNK(-4) }         // prev tile, rows h0-8..h0-1
    NCC_CHUNK(0) NCC_CHUNK(4) NCC_CHUNK(8) NCC_CHUNK(12) // center tile
    if (ht < 15) { NCC_CHUNK(16) NCC_CHUNK(20) }         // next tile, rows h0+16..h0+23
#undef NCC_CHUNK

    // C/D layout: vgpr r holds row (r + 8*half), column m.
#pragma unroll
    for (int r = 0; r < 8; ++r)
        ob[r * W_] = acc[r];
}

// ---------------------------------------------------------------------------
// Pass 3: D-axis running window + cc math + reduction. One thread per (h,w)
// column segment of DCH depths. Per-plane base pointer => five field loads are
// immediate offsets {0,256K,512K,768K,1M}. Trailing-edge re-reads hit L2.
// ---------------------------------------------------------------------------
__global__ void ncc_dpass_reduce(const float* __restrict__ in5, float* __restrict__ out) {
    const int chunk = blockIdx.x >> 8;                   // 0..3
    const int cb    = blockIdx.x & 255;
    const int col   = cb * 256 + threadIdx.x;            // 0..65535
    const int c0    = chunk * DCH;
    const int c1    = c0 + DCH;

    float r0 = 0.f, r1 = 0.f, r2 = 0.f, r3 = 0.f, r4 = 0.f;
#pragma unroll
    for (int d = c0 - PAD; d < c0 + PAD; ++d) {          // planes c0-5 .. c0+4
        if (0 <= d && d < D_) {
            const float* q = in5 + (size_t)d * (5 * HW_) + col;
            r0 += q[0 * HW_]; r1 += q[1 * HW_]; r2 += q[2 * HW_];
            r3 += q[3 * HW_]; r4 += q[4 * HW_];
        }
    }

    float acc = 0.f;
    for (int dout = c0; dout < c1; ++dout) {
        int da = dout + PAD;
        if (da < D_) {
            const float* q = in5 + (size_t)da * (5 * HW_) + col;
            r0 += q[0 * HW_]; r1 += q[1 * HW_]; r2 += q[2 * HW_];
            r3 += q[3 * HW_]; r4 += q[4 * HW_];
            if (da + 1 < D_)  // pull next leading plane toward the caches early
                __builtin_prefetch(q + 5 * HW_, 0, 1);
        }
        int ds = dout - PAD - 1;
        if (ds >= 0) {
            const float* q = in5 + (size_t)ds * (5 * HW_) + col;
            r0 -= q[0 * HW_]; r1 -= q[1 * HW_]; r2 -= q[2 * HW_];
            r3 -= q[3 * HW_]; r4 -= q[4 * HW_];
        }
        float cross = fmaxf(r4 - r0 * r1 * INV_VOL, EPS_);
        float ivar  = fmaxf(r2 - r0 * r0 * INV_VOL, EPS_);
        float jvar  = fmaxf(r3 - r1 * r1 * INV_VOL, EPS_);
        acc += cross / sqrtf(ivar * jvar);
    }

    __shared__ float red[256];
    red[threadIdx.x] = acc;
    __syncthreads();
#pragma unroll
    for (int s = 128; s > 0; s >>= 1) {
        if (threadIdx.x < s) red[threadIdx.x] += red[threadIdx.x + s];
        __syncthreads();
    }
    if (threadIdx.x == 0)
        atomicAdd(out, -red[0] * (1.0f / (float)NVOX));
}

// ---------------------------------------------------------------------------
extern "C" void kernel_launch(void* const* d_in, const int* in_sizes, int n_in,
                              void* d_out, int out_size, void* d_ws, size_t ws_size,
                              hipStream_t stream) {
    const float* I = (const float*)d_in[0];
    const float* J = (const float*)d_in[1];
    float* out  = (float*)d_out;
    float* buf0 = (float*)d_ws;                 // 5 * NVOX floats, [d][field][h][w]
    float* buf1 = buf0 + 5ull * NVOX;           // 5 * NVOX floats, [d][field][h][w]

    ncc_wpass<<<D_ * H_, W_, 0, stream>>>(I, J, buf0, out);
    ncc_hpass_wmma<<<(5 * D_ * 16 * 16) / 8, 256, 0, stream>>>(buf0, buf1);
    ncc_dpass_reduce<<<(HW_ / 256) * (D_ / DCH), 256, 0, stream>>>(buf1, out);
}